// ArcModel2Phase_30442728194558
// MI455X (gfx1250) — compile-verified
//
#include <hip/hip_runtime.h>
#include <math.h>

// ---------------------------------------------------------------------------
// ArcModel2Phase loss on gfx1250 (MI455X).
//
// log_integrand(i,j) = P'_i + Q'_j + x_i*txs_j + y_i*zg_j + log1p(-exp(-2*z))
//   z     = y_i * zg_j
//   txs_j = tx_j / sn^2,  zg_j = 2*G_j / sn^2
// Bilinear part computed with V_WMMA_F32_16X16X4_F32 (K=4 rank-4 form).
// ---------------------------------------------------------------------------

typedef float v2f __attribute__((ext_vector_type(2)));
typedef float v8f __attribute__((ext_vector_type(8)));

#define LOG2PI_F 1.8378770664093453f
#define WF_F     2.5f

// ws float layout:
//  [0..255]   zg[j]
//  [256..511] txs[j]
//  [512..767] Qp[j]
//  [768..783] scalars: lw0,lw1,lw2,C,ldc,intC,invsn2,inv2sn2,I1,I2
//  [1024 ..]  per-block partial sums

__global__ void arc_precompute_kernel(const float* __restrict__ k_u,
                                      const float* __restrict__ psb,
                                      const float* __restrict__ psn,
                                      const float* __restrict__ pI1,
                                      const float* __restrict__ pI2,
                                      const float* __restrict__ pw1,
                                      const float* __restrict__ pw2,
                                      const float* __restrict__ pw12,
                                      float* __restrict__ ws, int nmc) {
  const int j = threadIdx.x;
  const float sb = *psb, sn = *psn, I1 = *pI1, I2 = *pI2;
  const float dI = I2 - I1;
  const float invsn2 = 1.0f / (sn * sn);
  const float inv2sn2 = 0.5f * invsn2;
  const float erfW = erff(WF_F * 0.70710678118654752440f);
  const float I_min = I1 + 0.5f * dI * (1.0f - erfW);
  const float I_diff = dI * erfW;

  if (j < nmc) {
    const float k = k_u[j];
    const float tx = k * I_diff + I_min;
    const float u = 2.0f * (tx - I1) / dI - 1.0f;
    const float ei = erfinvf(u);
    const float ei2 = ei * ei;
    const float G = dI * (1.0f / sqrtf(2.0f * (float)M_PI * sb * sb)) * expf(-ei2);
    const float zg = 2.0f * G * invsn2;
    const float txs = tx * invsn2;
    const float Qp = ei2 - logf(G) - G * G * invsn2 - tx * tx * inv2sn2;
    ws[j]       = zg;
    ws[256 + j] = txs;
    ws[512 + j] = Qp;
  }
  if (j == 0) {
    const float r0 = *pw1, r1 = *pw2, r2 = *pw12;
    const float mx = fmaxf(r0, fmaxf(r1, r2));
    const float lse = mx + logf(expf(r0 - mx) + expf(r1 - mx) + expf(r2 - mx));
    ws[768] = r0 - lse;
    ws[769] = r1 - lse;
    ws[770] = r2 - lse;
    // Global constant C = -2 log sn - 0.5 log(pi) - log(2*WF*(I2-I1))
    ws[771] = -2.0f * logf(sn) - 0.5f * logf((float)M_PI) - logf(2.0f * WF_F * dI);
    // log(I_diff) - log(N_MC)
    ws[772] = logf(I_diff) - logf((float)nmc);
    // Interior constant: log2 - log(Gamma(1.5)) - 4 log sn - 0.5*LOG2PI
    ws[773] = logf(2.0f) - logf(0.88622692545275801365f) - 4.0f * logf(sn)
              - 0.5f * LOG2PI_F;
    ws[774] = invsn2;
    ws[775] = inv2sn2;
    ws[776] = I1;
    ws[777] = I2;
  }
}

// One wave handles a 16-row tile of points, looping over njt 16-wide j-tiles.
// 8 waves (256 threads) per block -> 128 points per block.
__global__ void __launch_bounds__(256)
arc_main_kernel(const float* __restrict__ X, const float* __restrict__ Y,
                const float* __restrict__ ws, float* __restrict__ partials,
                int M, int njt) {
  __shared__ float szg[256], stxs[256], sQp[256];
  __shared__ float wpart[8];

  const int t = threadIdx.x;
  // Stage per-j tables into LDS (3 KB).
  szg[t]  = ws[t];
  stxs[t] = ws[256 + t];
  sQp[t]  = ws[512 + t];
  const float lw0 = ws[768], lw1 = ws[769], lw2 = ws[770];
  const float Cc = ws[771], ldc = ws[772], intC = ws[773];
  const float invsn2 = ws[774], inv2sn2 = ws[775], I1v = ws[776], I2v = ws[777];
  __syncthreads();

  const int wave = t >> 5;
  const int lane = t & 31;
  const int li = lane & 15;
  const bool lohalf = lane < 16;
  const int i0 = (blockIdx.x * 8 + wave) * 16;
  const int idx = i0 + li;
  const bool valid = idx < M;
  const int cidx = valid ? idx : 0;

  const float xv = X[cidx];
  const float yv = Y[cidx];
  const float lgy = __logf(yv);
  const float Pp = Cc + lgy - yv * yv * invsn2 - xv * xv * inv2sn2;

  // A-matrix (16x4): rows = points, K = [P', x, y, 1].
  // Layout: VGPR v, lanes 0-15 -> A[lane][v]; lanes 16-31 -> A[lane-16][v+2].
  v2f a;
  a.x = lohalf ? Pp : yv;    // K0 = P' | K2 = y
  a.y = lohalf ? xv : 1.0f;  // K1 = x  | K3 = 1

  // y broadcast per output row r: D VGPR r holds rows i0+r (lo) / i0+r+8 (hi).
  float yr[8];
#pragma unroll
  for (int r = 0; r < 8; ++r) {
    const float ya = __shfl(yv, r, 32);
    const float yb = __shfl(yv, r + 8, 32);
    yr[r] = lohalf ? ya : yb;
  }

  // Streaming logsumexp state per D row.
  float m[8], s[8];
#pragma unroll
  for (int r = 0; r < 8; ++r) { m[r] = -1e30f; s[r] = 0.0f; }

  for (int jt = 0; jt < njt; ++jt) {
    const int jj = (jt << 4) + li;
    const float zgv = szg[jj];
    const float txsv = stxs[jj];
    const float qpv = sQp[jj];

    // B-matrix (4x16): cols = MC components, K = [1, txs, zg, Q'].
    // Layout: VGPR v, lanes 0-15 -> B[v][lane]; lanes 16-31 -> B[v+2][lane-16].
    v2f b;
    b.x = lohalf ? 1.0f : zgv;  // K0 = 1   | K2 = zg
    b.y = lohalf ? txsv : qpv;  // K1 = txs | K3 = Q'

    v8f c = {0.0f, 0.0f, 0.0f, 0.0f, 0.0f, 0.0f, 0.0f, 0.0f};
    // D[i][j] = P'_i + Q'_j + x_i*txs_j + y_i*zg_j  (the entire linear part)
    v8f d = __builtin_amdgcn_wmma_f32_16x16x4_f32(
        false, a, false, b, (short)0, c, false, false);

#pragma unroll
    for (int r = 0; r < 8; ++r) {
      const float z = yr[r] * zgv;              // z = 2 y G / sn^2  (>= ~2 here)
      const float e = __expf(-2.0f * z);
      // log1p(-e), e <= ~0.015 -> cubic series is exact to float precision
      const float l1 = -e * (1.0f + e * (0.5f + 0.333333333f * e));
      const float v = d[r] + l1;
      const float nm = fmaxf(m[r], v);
      s[r] = s[r] * __expf(m[r] - nm) + __expf(v - nm);
      m[r] = nm;
    }
  }

  // Combine LSE across the 16 lanes of each half (j runs over lanes too).
  float val[8];
#pragma unroll
  for (int r = 0; r < 8; ++r) {
    float mm = m[r], ss = s[r];
#pragma unroll
    for (int msk = 1; msk < 16; msk <<= 1) {
      const float mo = __shfl_xor(mm, msk, 32);
      const float so = __shfl_xor(ss, msk, 32);
      const float nm = fmaxf(mm, mo);
      ss = ss * __expf(mm - nm) + so * __expf(mo - nm);
      mm = nm;
    }
    val[r] = mm + __logf(ss);
  }

  // Redistribute: lane l (point i0 + (l&15)) picks the LSE for its own row.
  // val[r] holds row i0+r in lanes 0-15, row i0+r+8 in lanes 16-31.
  const bool own = (((lane >> 3) & 1) == ((lane >> 4) & 1));
  const int rsel = lane & 7;
  float p12v = 0.0f;
#pragma unroll
  for (int r = 0; r < 8; ++r) {
    const float other = __shfl_xor(val[r], 16, 32);
    const float pick = own ? val[r] : other;
    if (rsel == r) p12v = pick;
  }
  const float log_p12 = ldc + p12v;

  // Interior components + final 3-way logsumexp with mixture weights.
  const float common = intC + 2.0f * lgy - yv * yv * invsn2;
  const float d1 = xv - I1v, d2 = xv - I2v;
  const float lp1 = common - d1 * d1 * inv2sn2;
  const float lp2 = common - d2 * d2 * inv2sn2;
  const float a0 = lw0 + lp1, a1 = lw1 + lp2, a2 = lw2 + log_p12;
  const float mx3 = fmaxf(a0, fmaxf(a1, a2));
  const float lp = mx3 + __logf(__expf(a0 - mx3) + __expf(a1 - mx3) + __expf(a2 - mx3));

  // Each point is covered by both lane-halves; only the low half contributes.
  float contrib = (lohalf && valid) ? -lp : 0.0f;
#pragma unroll
  for (int msk = 1; msk < 32; msk <<= 1) contrib += __shfl_xor(contrib, msk, 32);
  if (lane == 0) wpart[wave] = contrib;
  __syncthreads();
  if (t == 0) {
    float sb = 0.0f;
#pragma unroll
    for (int w = 0; w < 8; ++w) sb += wpart[w];
    partials[blockIdx.x] = sb;
  }
}

// Deterministic final reduction: fixed traversal order, single block.
__global__ void arc_reduce_kernel(const float* __restrict__ partials, int n,
                                  float* __restrict__ out) {
  __shared__ float sm[256];
  const int t = threadIdx.x;
  float acc = 0.0f;
  for (int p = t; p < n; p += 256) acc += partials[p];
  sm[t] = acc;
  __syncthreads();
  for (int stride = 128; stride > 0; stride >>= 1) {
    if (t < stride) sm[t] += sm[t + stride];
    __syncthreads();
  }
  if (t == 0) out[0] = sm[0];
}

extern "C" void kernel_launch(void* const* d_in, const int* in_sizes, int n_in,
                              void* d_out, int out_size, void* d_ws, size_t ws_size,
                              hipStream_t stream) {
  const float* X   = (const float*)d_in[0];
  const float* Y   = (const float*)d_in[1];
  const float* KU  = (const float*)d_in[2];
  const float* SB  = (const float*)d_in[3];
  const float* SN  = (const float*)d_in[4];
  const float* I1  = (const float*)d_in[5];
  const float* I2  = (const float*)d_in[6];
  const float* W1  = (const float*)d_in[7];
  const float* W2  = (const float*)d_in[8];
  const float* W12 = (const float*)d_in[9];

  const int M = in_sizes[0];
  const int nmc = in_sizes[2];  // 256

  float* ws = (float*)d_ws;
  float* partials = ws + 1024;

  arc_precompute_kernel<<<1, nmc, 0, stream>>>(KU, SB, SN, I1, I2, W1, W2, W12,
                                               ws, nmc);

  const int tiles = (M + 15) / 16;
  const int blocks = (tiles + 7) / 8;
  arc_main_kernel<<<blocks, 256, 0, stream>>>(X, Y, ws, partials, M, nmc / 16);

  arc_reduce_kernel<<<1, 256, 0, stream>>>(partials, blocks, (float*)d_out);
}